// Model_86174223827142
// MI455X (gfx1250) — compile-verified
//
#include <hip/hip_runtime.h>
#include <hip/hip_bf16.h>

#define HDIM 128

typedef __attribute__((ext_vector_type(2))) float v2f;
typedef __attribute__((ext_vector_type(8))) float v8f;

// ---------------------------------------------------------------------------
// zero a float buffer
__global__ void k_zero(float* __restrict__ p, size_t n) {
    size_t t = (size_t)blockIdx.x * blockDim.x + threadIdx.x;
    if (t < n) p[t] = 0.0f;
}

// ---------------------------------------------------------------------------
// degree count: deg[idx[e]] += 1  (float atomics; accumulator is L2-resident)
__global__ void k_count_deg(const int* __restrict__ idx, float* __restrict__ deg, int ne) {
    int t = blockIdx.x * blockDim.x + threadIdx.x;
    if (t < ne) unsafeAtomicAdd(&deg[idx[t]], 1.0f);
}

// deg -> 1/max(deg,1)
__global__ void k_to_inv(float* __restrict__ deg, int n) {
    int t = blockIdx.x * blockDim.x + threadIdx.x;
    if (t < n) deg[t] = 1.0f / fmaxf(deg[t], 1.0f);
}

// ---------------------------------------------------------------------------
// x_dis[i,:] = dis_x[i] * lin_w[:] + lin_b[:] + dis_emb[i,:]
// one thread per float4 chunk (32 chunks per 128-wide row)
__global__ void k_build_xdis(const float* __restrict__ dis_x,
                             const float* __restrict__ lin_w,
                             const float* __restrict__ lin_b,
                             const float* __restrict__ dis_emb,
                             float* __restrict__ xdis, int n) {
    size_t t = (size_t)blockIdx.x * blockDim.x + threadIdx.x;
    size_t total = (size_t)n * (HDIM / 4);
    if (t >= total) return;
    int i = (int)(t >> 5);
    int c = (int)(t & 31);
    float xv = dis_x[i];
    float4 w = *(const float4*)(lin_w + c * 4);
    float4 b = *(const float4*)(lin_b + c * 4);
    float4 e = *(const float4*)(dis_emb + (size_t)i * HDIM + c * 4);
    float4 r;
    r.x = fmaf(xv, w.x, b.x) + e.x;
    r.y = fmaf(xv, w.y, b.y) + e.y;
    r.z = fmaf(xv, w.z, b.z) + e.z;
    r.w = fmaf(xv, w.w, b.w) + e.w;
    *(float4*)(xdis + (size_t)i * HDIM + c * 4) = r;
}

// ---------------------------------------------------------------------------
// scatter-add: agg[dst[e], :] += xsrc[src[e], :]
// one wave (32 lanes) per edge; each lane owns a float4 chunk of the row.
__global__ void k_scatter_add(const float* __restrict__ xsrc,
                              const int* __restrict__ esrc,
                              const int* __restrict__ edst,
                              float* __restrict__ agg, int ne) {
    size_t t = (size_t)blockIdx.x * blockDim.x + threadIdx.x;
    size_t total = (size_t)ne * 32;
    if (t >= total) return;
    int e = (int)(t >> 5);
    int c = (int)(t & 31);
    int s = esrc[e];
    int d = edst[e];
    float4 v = *(const float4*)(xsrc + (size_t)s * HDIM + c * 4);
    float* dp = agg + (size_t)d * HDIM + c * 4;
    unsafeAtomicAdd(dp + 0, v.x);
    unsafeAtomicAdd(dp + 1, v.y);
    unsafeAtomicAdd(dp + 2, v.z);
    unsafeAtomicAdd(dp + 3, v.w);
}

// ---------------------------------------------------------------------------
// Fused SAGE transform:
//   out[i,:] = (agg[i,:] * invc[i]) @ Wl^T + xdst[i,:] @ Wr^T + bl   (+ReLU)
// Block = 256 threads = 8 wave32s; block covers 16 rows x 128 cols.
// Wave w computes the 16-column N-tile [16w,16w+16) via V_WMMA_F32_16X16X4_F32,
// accumulating both matrix products into one C fragment over K=128.
__global__ void __launch_bounds__(256)
k_sage_gemm(const float* __restrict__ agg,   // [n,H] summed neighbor feats
            const float* __restrict__ invc,  // [n] 1/max(deg,1)
            const float* __restrict__ xdst,  // [n,H]
            const float* __restrict__ Wl,    // [H,H] (h_out, k)
            const float* __restrict__ bl,    // [H]
            const float* __restrict__ Wr,    // [H,H]
            float* __restrict__ out,         // [n,H]
            int n, int do_relu) {
    const int row0  = blockIdx.x * 16;
    const int wave  = threadIdx.x >> 5;  // 0..7 -> N tile
    const int lane  = threadIdx.x & 31;
    const int m     = lane & 15;         // A row / B col / D col within tile
    const int khalf = lane >> 4;         // 0: K+{0,1}, 1: K+{2,3}

    const int   r      = row0 + m;       // n is a multiple of 16 (50000)
    const float ic     = invc[r];
    const float* aggRow = agg  + (size_t)r * HDIM;
    const float* xRow   = xdst + (size_t)r * HDIM;
    const int    nc     = wave * 16 + m; // output column this lane owns in B/D
    const float* wlRow  = Wl + (size_t)nc * HDIM;
    const float* wrRow  = Wr + (size_t)nc * HDIM;

    v8f acc = {};
#pragma unroll
    for (int kb = 0; kb < HDIM; kb += 4) {
        const int ko = kb + 2 * khalf;
        v2f a1, a2, b1, b2;
        // A fragment (16x4): lane pair K = ko, ko+1 from its activation row
        a1.x = aggRow[ko] * ic;  a1.y = aggRow[ko + 1] * ic;
        a2.x = xRow[ko];         a2.y = xRow[ko + 1];
        // B fragment (4x16): B[k][n] = W[n][k] -> contiguous from W row nc
        b1.x = wlRow[ko];        b1.y = wlRow[ko + 1];
        b2.x = wrRow[ko];        b2.y = wrRow[ko + 1];
        acc = __builtin_amdgcn_wmma_f32_16x16x4_f32(
            false, a1, false, b1, (short)0, acc, false, false);
        acc = __builtin_amdgcn_wmma_f32_16x16x4_f32(
            false, a2, false, b2, (short)0, acc, false, false);
    }

    const float bias = bl[nc];
#pragma unroll
    for (int j = 0; j < 8; ++j) {         // D VGPR j -> row M = j + 8*khalf
        const int rr = row0 + j + 8 * khalf;
        float v = acc[j] + bias;
        if (do_relu) v = v > 0.0f ? v : 0.0f;
        out[(size_t)rr * HDIM + nc] = v;
    }
}

// ---------------------------------------------------------------------------
// edge scoring: out[e] = dot(o_drug[ls[e]], o_dis[ld[e]]) ; one wave per edge
__global__ void k_edge_score(const float* __restrict__ od,
                             const float* __restrict__ ods,
                             const int* __restrict__ ls,
                             const int* __restrict__ ld,
                             float* __restrict__ out, int ne) {
    int e    = blockIdx.x * (blockDim.x >> 5) + (threadIdx.x >> 5);
    int lane = threadIdx.x & 31;
    if (e >= ne) return;
    const float4 a = *(const float4*)(od  + (size_t)ls[e] * HDIM + lane * 4);
    const float4 b = *(const float4*)(ods + (size_t)ld[e] * HDIM + lane * 4);
    float s = a.x * b.x + a.y * b.y + a.z * b.z + a.w * b.w;
#pragma unroll
    for (int off = 16; off > 0; off >>= 1) s += __shfl_xor(s, off, 32);
    if (lane == 0) out[e] = s;
}

// ---------------------------------------------------------------------------
extern "C" void kernel_launch(void* const* d_in, const int* in_sizes, int n_in,
                              void* d_out, int out_size, void* d_ws, size_t ws_size,
                              hipStream_t stream) {
    const float* dis_x    = (const float*)d_in[0];
    const float* drug_emb = (const float*)d_in[1];
    const float* dis_emb  = (const float*)d_in[2];
    const float* lin_w    = (const float*)d_in[3];
    const float* lin_b    = (const float*)d_in[4];
    const float* Wl       = (const float*)d_in[5];   // [4,H,H]
    const float* bl       = (const float*)d_in[6];   // [4,H]
    const float* Wr       = (const float*)d_in[7];   // [4,H,H]
    const int*   e_src    = (const int*)d_in[10];
    const int*   e_dst    = (const int*)d_in[11];
    const int*   el_src   = (const int*)d_in[12];
    const int*   el_dst   = (const int*)d_in[13];
    float*       outp     = (float*)d_out;

    const int n_dis  = in_sizes[0];
    const int H      = in_sizes[4];        // 128
    const int n_drug = in_sizes[1] / H;
    const int E      = in_sizes[10];
    const int EL     = in_sizes[12];

    const size_t NFdis  = (size_t)n_dis * HDIM;
    const size_t NFdrug = (size_t)n_drug * HDIM;
    const size_t NFmax  = NFdis > NFdrug ? NFdis : NFdrug;

    float* ws = (float*)d_ws;
    float* xdis    = ws;                 ws += NFdis;
    float* hdrug   = ws;                 ws += NFdrug;
    float* hdis    = ws;                 ws += NFdis;
    float* odrug   = ws;                 ws += NFdrug;
    float* odis    = ws;                 ws += NFdis;
    float* aggbuf  = ws;                 ws += NFmax;
    float* inv_drug = ws;                ws += (size_t)n_drug;
    float* inv_dis  = ws;                ws += (size_t)n_dis;

    const int B = 256;
    auto blocksF = [&](size_t n) { return (unsigned)((n + B - 1) / B); };

    // degrees -> reciprocal
    k_zero<<<blocksF(n_drug), B, 0, stream>>>(inv_drug, (size_t)n_drug);
    k_zero<<<blocksF(n_dis),  B, 0, stream>>>(inv_dis,  (size_t)n_dis);
    k_count_deg<<<blocksF(E), B, 0, stream>>>(e_src, inv_drug, E);
    k_count_deg<<<blocksF(E), B, 0, stream>>>(e_dst, inv_dis,  E);
    k_to_inv<<<blocksF(n_drug), B, 0, stream>>>(inv_drug, n_drug);
    k_to_inv<<<blocksF(n_dis),  B, 0, stream>>>(inv_dis,  n_dis);

    // x_dis = dis_x * lin_w^T + lin_b + dis_emb   (x_drug == drug_emb: id gather)
    k_build_xdis<<<blocksF((size_t)n_dis * 32), B, 0, stream>>>(
        dis_x, lin_w, lin_b, dis_emb, xdis, n_dis);

    const size_t scatT = (size_t)E * 32;
    const unsigned gDrug = (unsigned)(n_drug / 16);
    const unsigned gDis  = (unsigned)(n_dis / 16);

    // layer 1: drug side  (msg = x_dis[e_dst] -> e_src)
    k_zero<<<blocksF(NFdrug), B, 0, stream>>>(aggbuf, NFdrug);
    k_scatter_add<<<blocksF(scatT), B, 0, stream>>>(xdis, e_dst, e_src, aggbuf, E);
    k_sage_gemm<<<gDrug, B, 0, stream>>>(aggbuf, inv_drug, drug_emb,
                                         Wl + 0 * H * H, bl + 0 * H, Wr + 0 * H * H,
                                         hdrug, n_drug, 1);
    // layer 1: disease side (msg = x_drug[e_src] -> e_dst)
    k_zero<<<blocksF(NFdis), B, 0, stream>>>(aggbuf, NFdis);
    k_scatter_add<<<blocksF(scatT), B, 0, stream>>>(drug_emb, e_src, e_dst, aggbuf, E);
    k_sage_gemm<<<gDis, B, 0, stream>>>(aggbuf, inv_dis, xdis,
                                        Wl + 1 * H * H, bl + 1 * H, Wr + 1 * H * H,
                                        hdis, n_dis, 1);
    // layer 2: drug side (no relu)
    k_zero<<<blocksF(NFdrug), B, 0, stream>>>(aggbuf, NFdrug);
    k_scatter_add<<<blocksF(scatT), B, 0, stream>>>(hdis, e_dst, e_src, aggbuf, E);
    k_sage_gemm<<<gDrug, B, 0, stream>>>(aggbuf, inv_drug, hdrug,
                                         Wl + 2 * H * H, bl + 2 * H, Wr + 2 * H * H,
                                         odrug, n_drug, 0);
    // layer 2: disease side (no relu)
    k_zero<<<blocksF(NFdis), B, 0, stream>>>(aggbuf, NFdis);
    k_scatter_add<<<blocksF(scatT), B, 0, stream>>>(hdrug, e_src, e_dst, aggbuf, E);
    k_sage_gemm<<<gDis, B, 0, stream>>>(aggbuf, inv_dis, hdis,
                                        Wl + 3 * H * H, bl + 3 * H, Wr + 3 * H * H,
                                        odis, n_dis, 0);

    // edge scoring: one wave per edge, 8 waves per block
    unsigned gScore = (unsigned)((EL + 7) / 8);
    k_edge_score<<<gScore, B, 0, stream>>>(odrug, odis, el_src, el_dst, outp, EL);
}